// _ECELoss_1357209665663
// MI455X (gfx1250) — compile-verified
//
#include <hip/hip_runtime.h>
#include <hip/hip_bf16.h>
#include <math.h>

// ECE loss, MI455X (gfx1250, wave32).
// Memory-bound: 512MB logits -> ~22us floor at 23.3 TB/s.
// One wave = one row (32 lanes x float4 = 128 f32, coalesced B128 NT loads).
// Wave reductions: pure-VALU DPP/permlane butterflies (foldable into v_max_dpp).
// Histogram accumulation: V_WMMA_F32_16X16X4_F32 (exact f32):
//   D(16x16) += OneHot(16bins x 4rows) * Stats(4rows x 16cols), kept in VGPRs.

typedef __attribute__((ext_vector_type(2))) float v2f;
typedef __attribute__((ext_vector_type(4))) float v4f;
typedef __attribute__((ext_vector_type(8))) float v8f;

#define NBINS 15

#if __has_builtin(__builtin_amdgcn_update_dpp) && \
    __has_builtin(__builtin_amdgcn_mov_dpp8) && \
    __has_builtin(__builtin_amdgcn_permlanex16)
#define ECE_USE_DPP 1
#else
#define ECE_USE_DPP 0
#endif

// Butterfly exchange x -> lane^step, wave32, all lanes active.
// old=0 + bound_ctrl=1 so GCNDPPCombine can fold the mov_dpp into the consumer.
template <int STEP>
__device__ __forceinline__ float bfly(float x) {
#if ECE_USE_DPP
    const int xi = __builtin_bit_cast(int, x);
    int r;
    if constexpr (STEP == 1) {
        r = __builtin_amdgcn_update_dpp(0, xi, 0xB1, 0xF, 0xF, true);   // quad_perm [1,0,3,2]
    } else if constexpr (STEP == 2) {
        r = __builtin_amdgcn_update_dpp(0, xi, 0x4E, 0xF, 0xF, true);   // quad_perm [2,3,0,1]
    } else if constexpr (STEP == 4) {
        r = __builtin_amdgcn_mov_dpp8(
            xi, (4 | (5 << 3) | (6 << 6) | (7 << 9) |
                 (0 << 12) | (1 << 15) | (2 << 18) | (3 << 21)));       // dpp8 xor4
    } else if constexpr (STEP == 8) {
        r = __builtin_amdgcn_update_dpp(0, xi, 0x128, 0xF, 0xF, true);  // row_ror:8 == xor8
    } else {
        r = __builtin_amdgcn_permlanex16(xi, xi, 0x76543210, 0xFEDCBA98,
                                         false, false);                 // cross-row == xor16
    }
    return __builtin_bit_cast(float, r);
#else
    return __shfl_xor(x, STEP, 32);
#endif
}

__device__ __forceinline__ float wave_max(float m) {
    m = fmaxf(m, bfly<1>(m));
    m = fmaxf(m, bfly<2>(m));
    m = fmaxf(m, bfly<4>(m));
    m = fmaxf(m, bfly<8>(m));
    m = fmaxf(m, bfly<16>(m));
    return m;
}

__device__ __forceinline__ float wave_sum(float s) {
    s += bfly<1>(s);
    s += bfly<2>(s);
    s += bfly<4>(s);
    s += bfly<8>(s);
    s += bfly<16>(s);
    return s;
}

// Process one row; returns conf, acc, bin (all lanes hold the results).
__device__ __forceinline__ void do_row(const float* __restrict__ logits,
                                       const int* __restrict__ labels,
                                       float inv_t, int r, int lane,
                                       float& conf, float& acc, int& bin) {
    const v4f v = __builtin_nontemporal_load(
        (const v4f*)(logits + (size_t)r * 128) + lane);
    float m = fmaxf(fmaxf(v.x, v.y), fmaxf(v.z, v.w));
    m = wave_max(m);  // all lanes hold row max
    // conf = softmax max prob = 1 / sum(exp((l-m)/t))
    float s = __expf((v.x - m) * inv_t) + __expf((v.y - m) * inv_t) +
              __expf((v.z - m) * inv_t) + __expf((v.w - m) * inv_t);
    s = wave_sum(s);
    conf = 1.0f / s;
    // prediction == label  <=>  logits[label] == rowmax (ties measure-zero)
    const int l = labels[r];  // wave-uniform -> scalar load
    const int le = l & 3;
    const float lv = (le == 0) ? v.x : (le == 1) ? v.y : (le == 2) ? v.z : v.w;
    const bool hit = (lane == (l >> 2)) && (lv == m);
    acc = (__ballot(hit) != 0ULL) ? 1.0f : 0.0f;
    int b = (int)ceilf(conf * 15.0f) - 1;  // conf in (b/15,(b+1)/15]
    bin = b < 0 ? 0 : (b > 14 ? 14 : b);
}

// Fold 4 rows' {count, conf, acc} into the WMMA accumulator.
__device__ __forceinline__ void wmma_accum(int lane,
                                           const float conf[4], const float accv[4],
                                           const float cntv[4], const int binv[4],
                                           v8f& accD) {
    // A 16x4 (M=bin, K=row): lanes 0-15 hold K=0 (v0) / K=1 (v1); lanes 16-31 K=2/K=3.
    // B 4x16 (K=row, N=stat): mirrored K split across lane halves; N = lane&15.
    const bool hi = lane >= 16;
    const int  mn = lane & 15;

    const int   binA = hi ? binv[2] : binv[0];
    const int   binB = hi ? binv[3] : binv[1];
    const float cntA = hi ? cntv[2] : cntv[0];
    const float cntB = hi ? cntv[3] : cntv[1];
    const float cfA  = hi ? conf[2] : conf[0];
    const float cfB  = hi ? conf[3] : conf[1];
    const float acA  = hi ? accv[2] : accv[0];
    const float acB  = hi ? accv[3] : accv[1];

    v2f A, B;
    A.x = (binA == mn) ? 1.0f : 0.0f;
    A.y = (binB == mn) ? 1.0f : 0.0f;
    B.x = (mn == 0) ? cntA : (mn == 1) ? cfA : (mn == 2) ? acA : 0.0f;
    B.y = (mn == 0) ? cntB : (mn == 1) ? cfB : (mn == 2) ? acB : 0.0f;

    accD = __builtin_amdgcn_wmma_f32_16x16x4_f32(
        false, A, false, B, (short)0, accD, false, false);
}

__global__ void ece_init(float* __restrict__ ws) {
    int i = threadIdx.x;
    if (i < 48) ws[i] = 0.0f;
}

__global__ __launch_bounds__(256) void ece_main(const float* __restrict__ logits,
                                                const int* __restrict__ labels,
                                                const float* __restrict__ tptr,
                                                float* __restrict__ ws,
                                                int nrows) {
    __shared__ float sbin[48];  // [3 stats][16 bins], bin 15 unused
    if (threadIdx.x < 48) sbin[threadIdx.x] = 0.0f;
    __syncthreads();

    const float inv_t = 1.0f / tptr[0];
    const int lane = threadIdx.x & 31;
    const int wave = threadIdx.x >> 5;
    const int wavesPerBlock = blockDim.x >> 5;
    const int gwave = blockIdx.x * wavesPerBlock + wave;
    const int nwaves = gridDim.x * wavesPerBlock;
    const int groupStride = nwaves * 4;

    v8f accD = {0.f, 0.f, 0.f, 0.f, 0.f, 0.f, 0.f, 0.f};

    float conf[4], accv[4], cntv[4];
    int binv[4];

    int base = gwave * 4;
    // Hot loop: full 4-row groups, no per-row guards, unguarded speculative prefetch.
    for (; base + 3 < nrows; base += groupStride) {
        __builtin_prefetch(logits + (size_t)(base + groupStride) * 128 + lane * 4, 0, 3);
#pragma unroll
        for (int k = 0; k < 4; ++k) {
            do_row(logits, labels, inv_t, base + k, lane, conf[k], accv[k], binv[k]);
            cntv[k] = 1.0f;
        }
        wmma_accum(lane, conf, accv, cntv, binv, accD);
    }
    // Tail: at most one partial group per wave.
    if (base < nrows) {
#pragma unroll
        for (int k = 0; k < 4; ++k) {
            if (base + k < nrows) {  // uniform across wave
                do_row(logits, labels, inv_t, base + k, lane, conf[k], accv[k], binv[k]);
                cntv[k] = 1.0f;
            } else {
                conf[k] = 0.0f; accv[k] = 0.0f; cntv[k] = 0.0f; binv[k] = -1;
            }
        }
        wmma_accum(lane, conf, accv, cntv, binv, accD);
    }

    // Drain accumulators: C/D layout: VGPR j, lanes 0-15 -> M=j, lanes 16-31 -> M=j+8;
    // N = lane&15. Stats live in N=0..2.
    {
        const int n = lane & 15;
        if (n <= 2) {
            const int mbase = (lane < 16) ? 0 : 8;
#pragma unroll
            for (int j = 0; j < 8; ++j) {
                const int m = mbase + j;
                if (m < NBINS) atomicAdd(&sbin[n * 16 + m], accD[j]);
            }
        }
    }
    __syncthreads();
    if (threadIdx.x < 48) {
        const int n2 = threadIdx.x >> 4;
        const int m2 = threadIdx.x & 15;
        if (m2 < NBINS) {
            const float v = sbin[threadIdx.x];
            if (v != 0.0f) atomicAdd(&ws[n2 * 16 + m2], v);
        }
    }
}

__global__ void ece_final(const float* __restrict__ ws, float* __restrict__ out, int nrows) {
    if (blockIdx.x == 0 && threadIdx.x == 0) {
        const float nf = (float)nrows;
        float ece = 0.0f;
        for (int k = 0; k < NBINS; ++k) {
            const float cnt = ws[0 * 16 + k];
            const float cs  = ws[1 * 16 + k];
            const float as  = ws[2 * 16 + k];
            if (cnt > 0.0f) {
                const float safe = fmaxf(cnt, 1.0f);
                const float gap  = fabsf(cs / safe - as / safe);
                ece += gap * (cnt / nf);
            }
        }
        out[0] = ece;
    }
}

extern "C" void kernel_launch(void* const* d_in, const int* in_sizes, int n_in,
                              void* d_out, int out_size, void* d_ws, size_t ws_size,
                              hipStream_t stream) {
    const float* logits = (const float*)d_in[0];
    const int*   labels = (const int*)d_in[1];
    const float* t      = (const float*)d_in[2];
    float*       out    = (float*)d_out;
    float*       ws     = (float*)d_ws;
    const int nrows = in_sizes[1];   // N = 1048576 (labels element count)

    ece_init<<<1, 64, 0, stream>>>(ws);
    ece_main<<<1024, 256, 0, stream>>>(logits, labels, t, ws, nrows);
    ece_final<<<1, 1, 0, stream>>>(ws, out, nrows);
}